// PPO_42769284333854
// MI455X (gfx1250) — compile-verified
//
#include <hip/hip_runtime.h>
#include <math.h>

// ---------------- problem constants ----------------
#define T_TOTAL  2097152
#define NB       512          // blocks for the scan kernels
#define NT       256          // threads per block (8 wave32 waves)
#define PT       16           // elements per thread
#define CHUNK    (NT * PT)    // 4096 elems per block; NB*CHUNK == T_TOTAL

#define GAMMA_F   0.99f
#define GL_F      (0.99f * 0.95f)
#define EPS_F     0.2f
#define LOG2PI_F  1.8378770664093453f

// ---------------- workspace layout (floats) ----------------
#define WS_AR   0                 // per-block composed map: ret A
#define WS_BR   (WS_AR + NB)      // ret B
#define WS_AA   (WS_BR + NB)      // adv A
#define WS_BA   (WS_AA + NB)      // adv B
#define WS_CR   (WS_BA + NB)      // per-block incoming carry: ret
#define WS_CA   (WS_CR + NB)      // adv
#define WS_SA   (WS_CA + NB)      // per-block stat partials
#define WS_SA2  (WS_SA + NB)
#define WS_SR   (WS_SA2 + NB)
#define WS_SR2  (WS_SR + NB)
#define WS_SCAL (WS_SR2 + NB)     // 4 scalars: mean_a, rstd_a, mean_r, rstd_r
#define WS_LACT (WS_SCAL + 4)     // per-block actor partials
#define WS_LCRI (WS_LACT + NB)    // per-block critic partials
// total = 12*NB + 4 = 6148 floats (~25 KB)

typedef __attribute__((ext_vector_type(2))) float v2f;
typedef __attribute__((ext_vector_type(8))) float v8f;

// ---------------- WMMA-based block reduction (256 -> 1) ----------------
// 256 -> 64 via LDS, then one V_WMMA_F32_16X16X4_F32 with B = ones collapses
// the 64-value A matrix (16x4) into 16 row sums replicated across columns.
// Per-lane sum of the 8 D VGPRs leaves rows 0-7 in lanes 0-15 and rows 8-15
// in lanes 16-31; lds[0]+lds[16] is the grand total.
__device__ __forceinline__ float block_reduce_wmma(float val, float* sred) {
    const int tid = threadIdx.x;
    sred[tid] = val;
    __syncthreads();
    if (tid < 64)
        sred[tid] = sred[tid] + sred[tid + 64] + sred[tid + 128] + sred[tid + 192];
    __syncthreads();
    if (tid < 32) {                       // one full wave -> EXEC all ones
        v2f a; a[0] = sred[tid]; a[1] = sred[tid + 32];
        v2f b; b[0] = 1.0f;      b[1] = 1.0f;
        v8f c = {};
        c = __builtin_amdgcn_wmma_f32_16x16x4_f32(
                /*neg_a=*/false, a, /*neg_b=*/false, b,
                /*c_mod=*/(short)0, c, /*reuse_a=*/false, /*reuse_b=*/false);
        sred[tid] = c[0] + c[1] + c[2] + c[3] + c[4] + c[5] + c[6] + c[7];
    }
    __syncthreads();
    float tot = sred[0] + sred[16];
    __syncthreads();
    return tot;
}

// ---------------- shared GAE machinery ----------------
// Loads this thread's PT-element chunk, composes its reverse affine map for
// both recurrences, does an LDS suffix scan across the 256 threads, and
// produces this thread's incoming carry (value of ret/adv at its right edge).
__device__ __forceinline__ void gae_scan_block(
    const float* __restrict__ rewards, const float* __restrict__ values,
    const int* __restrict__ masks, float Xr, float Xa,
    float r[PT], float v[PT], float m[PT],
    float& vnext_out, float& cr, float& ca,
    float* sAr, float* sBr, float* sAa, float* sBa)
{
    const int tid = threadIdx.x;
    const int s   = blockIdx.x * CHUNK + tid * PT;

    const float4* r4 = (const float4*)(rewards + s);
    const float4* v4 = (const float4*)(values + s);
    const int4*   m4 = (const int4*)(masks + s);
#pragma unroll
    for (int q = 0; q < PT / 4; ++q) {
        float4 rr = r4[q]; float4 vv = v4[q]; int4 mm = m4[q];
        r[4*q+0] = rr.x; r[4*q+1] = rr.y; r[4*q+2] = rr.z; r[4*q+3] = rr.w;
        v[4*q+0] = vv.x; v[4*q+1] = vv.y; v[4*q+2] = vv.z; v[4*q+3] = vv.w;
        m[4*q+0] = (float)mm.x; m[4*q+1] = (float)mm.y;
        m[4*q+2] = (float)mm.z; m[4*q+3] = (float)mm.w;
    }
    const int e = s + PT;
    const float vnext0 = (e == T_TOTAL) ? 0.0f : values[e];
    vnext_out = vnext0;

    // compose affine maps right-to-left: (A,B) <- (a*A, a*B + b)
    float Ar = 1.0f, Br = 0.0f, Aa = 1.0f, Ba = 0.0f;
    float vn = vnext0;
#pragma unroll
    for (int i = PT - 1; i >= 0; --i) {
        float a     = GAMMA_F * m[i];
        float al    = GL_F * m[i];
        float delta = fmaf(a, vn, r[i]) - v[i];
        Br = fmaf(a, Br, r[i]);   Ar *= a;
        Ba = fmaf(al, Ba, delta); Aa *= al;
        vn = v[i];
    }

    sAr[tid] = Ar; sBr[tid] = Br; sAa[tid] = Aa; sBa[tid] = Ba;
    __syncthreads();
    // inclusive suffix scan (Hillis-Steele) over affine maps
    for (int st = 1; st < NT; st <<= 1) {
        float a1 = sAr[tid], b1 = sBr[tid], a3 = sAa[tid], b3 = sBa[tid];
        float a2 = 1.0f, b2 = 0.0f, a4 = 1.0f, b4 = 0.0f;
        bool ok = (tid + st) < NT;
        if (ok) { a2 = sAr[tid+st]; b2 = sBr[tid+st]; a4 = sAa[tid+st]; b4 = sBa[tid+st]; }
        __syncthreads();
        if (ok) { sAr[tid] = a1*a2; sBr[tid] = fmaf(a1, b2, b1);
                  sAa[tid] = a3*a4; sBa[tid] = fmaf(a3, b4, b3); }
        __syncthreads();
    }
    // carry at this thread's right edge = suffix map of threads tid+1.. applied to X
    if (tid == NT - 1) { cr = Xr; ca = Xa; }
    else {
        cr = fmaf(sAr[tid+1], Xr, sBr[tid+1]);
        ca = fmaf(sAa[tid+1], Xa, sBa[tid+1]);
    }
}

// ---------------- K1: per-block composed maps ----------------
__global__ void k1_block_maps(const float* __restrict__ rw,
                              const float* __restrict__ vals,
                              const int* __restrict__ masks,
                              float* __restrict__ ws) {
    __shared__ float sAr[NT], sBr[NT], sAa[NT], sBa[NT];
    float r[PT], v[PT], m[PT], vn, cr, ca;
    gae_scan_block(rw, vals, masks, 0.0f, 0.0f, r, v, m, vn, cr, ca, sAr, sBr, sAa, sBa);
    if (threadIdx.x == 0) {
        const int b = blockIdx.x;
        ws[WS_AR + b] = sAr[0]; ws[WS_BR + b] = sBr[0];
        ws[WS_AA + b] = sAa[0]; ws[WS_BA + b] = sBa[0];
    }
}

// ---------------- K2: serial reverse scan over 512 block summaries ----------------
__global__ void k2_block_scan(float* __restrict__ ws) {
    if (threadIdx.x == 0 && blockIdx.x == 0) {
        float xr = 0.0f, xa = 0.0f;
        for (int b = NB - 1; b >= 0; --b) {
            ws[WS_CR + b] = xr; ws[WS_CA + b] = xa;
            xr = fmaf(ws[WS_AR + b], xr, ws[WS_BR + b]);
            xa = fmaf(ws[WS_AA + b], xa, ws[WS_BA + b]);
        }
    }
}

// ---------------- K3: emit rets/advs on the fly, accumulate stats ----------------
__global__ void k3_stats(const float* __restrict__ rw,
                         const float* __restrict__ vals,
                         const int* __restrict__ masks,
                         float* __restrict__ ws) {
    __shared__ float sAr[NT], sBr[NT], sAa[NT], sBa[NT];
    __shared__ float sred[NT];
    const int b = blockIdx.x;
    const float Xr = ws[WS_CR + b], Xa = ws[WS_CA + b];
    float r[PT], v[PT], m[PT], vn, cr, ca;
    gae_scan_block(rw, vals, masks, Xr, Xa, r, v, m, vn, cr, ca, sAr, sBr, sAa, sBa);

    float pr = cr, pa = ca, vv = vn;
    float sa = 0.f, sa2 = 0.f, sr = 0.f, sr2 = 0.f;
#pragma unroll
    for (int i = PT - 1; i >= 0; --i) {
        float mt    = m[i];
        float ret   = fmaf(GAMMA_F * mt, pr, r[i]);
        float delta = fmaf(GAMMA_F * mt, vv, r[i]) - v[i];
        float adv   = fmaf(GL_F * mt, pa, delta);
        sa += adv; sa2 = fmaf(adv, adv, sa2);
        sr += ret; sr2 = fmaf(ret, ret, sr2);
        pr = ret; pa = adv; vv = v[i];
    }
    sa  = block_reduce_wmma(sa,  sred);
    sa2 = block_reduce_wmma(sa2, sred);
    sr  = block_reduce_wmma(sr,  sred);
    sr2 = block_reduce_wmma(sr2, sred);
    if (threadIdx.x == 0) {
        ws[WS_SA + b] = sa;  ws[WS_SA2 + b] = sa2;
        ws[WS_SR + b] = sr;  ws[WS_SR2 + b] = sr2;
    }
}

// ---------------- K4: finalize mean / rstd ----------------
__global__ void k4_stats_final(float* __restrict__ ws) {
    __shared__ float sred[NT];
    const int tid = threadIdx.x;
    float sa  = ws[WS_SA  + tid] + ws[WS_SA  + tid + NT];
    float sa2 = ws[WS_SA2 + tid] + ws[WS_SA2 + tid + NT];
    float sr  = ws[WS_SR  + tid] + ws[WS_SR  + tid + NT];
    float sr2 = ws[WS_SR2 + tid] + ws[WS_SR2 + tid + NT];
    sa  = block_reduce_wmma(sa,  sred);
    sa2 = block_reduce_wmma(sa2, sred);
    sr  = block_reduce_wmma(sr,  sred);
    sr2 = block_reduce_wmma(sr2, sred);
    if (tid == 0) {
        const float N = (float)T_TOTAL;
        float mean_a = sa / N;
        float var_a  = fmaxf((sa2 - sa * sa / N) / (N - 1.0f), 0.0f);
        float mean_r = sr / N;
        float var_r  = fmaxf((sr2 - sr * sr / N) / (N - 1.0f), 0.0f);
        ws[WS_SCAL + 0] = mean_a;
        ws[WS_SCAL + 1] = 1.0f / (sqrtf(var_a) + 1e-10f);
        ws[WS_SCAL + 2] = mean_r;
        ws[WS_SCAL + 3] = 1.0f / (sqrtf(var_r) + 1e-10f);
    }
}

// ---------------- K5: recompute GAE, compute actor/critic partials ----------------
__global__ void k5_loss(const float* __restrict__ rw,
                        const float* __restrict__ vals,
                        const int* __restrict__ masks,
                        const float* __restrict__ beta,
                        const float* __restrict__ act,
                        const float* __restrict__ mu,
                        const float* __restrict__ lv,
                        float* __restrict__ ws) {
    __shared__ float sAr[NT], sBr[NT], sAa[NT], sBa[NT];
    __shared__ float sred[NT];
    const int b = blockIdx.x;
    const float Xr = ws[WS_CR + b], Xa = ws[WS_CA + b];
    const float mean_a = ws[WS_SCAL + 0], rstd_a = ws[WS_SCAL + 1];
    const float mean_r = ws[WS_SCAL + 2], rstd_r = ws[WS_SCAL + 3];

    float r[PT], v[PT], m[PT], vn, cr, ca;
    gae_scan_block(rw, vals, masks, Xr, Xa, r, v, m, vn, cr, ca, sAr, sBr, sAa, sBa);

    const int s = b * CHUNK + threadIdx.x * PT;
    const float4* b4 = (const float4*)beta;  // (T,4) rows -> one float4 each
    const float4* a4 = (const float4*)act;
    const float4* u4 = (const float4*)mu;
    const float4* l4 = (const float4*)lv;

    float pr = cr, pa = ca, vv = vn;
    float sAct = 0.f, sCri = 0.f;
#pragma unroll
    for (int i = PT - 1; i >= 0; --i) {
        float mt    = m[i];
        float ret   = fmaf(GAMMA_F * mt, pr, r[i]);
        float delta = fmaf(GAMMA_F * mt, vv, r[i]) - v[i];
        float adv   = fmaf(GL_F * mt, pa, delta);
        pr = ret; pa = adv; vv = v[i];

        float advn = (adv - mean_a) * rstd_a;
        float retn = (ret - mean_r) * rstd_r;

        const int t = s + i;
        float4 bb = b4[t], aa = a4[t], uu = u4[t], ll = l4[t];
        float lp = 0.0f, d;
        d = aa.x - uu.x; lp += -0.5f * d * d * expf(-ll.x) - 0.5f * ll.x - 0.5f * LOG2PI_F - bb.x;
        d = aa.y - uu.y; lp += -0.5f * d * d * expf(-ll.y) - 0.5f * ll.y - 0.5f * LOG2PI_F - bb.y;
        d = aa.z - uu.z; lp += -0.5f * d * d * expf(-ll.z) - 0.5f * ll.z - 0.5f * LOG2PI_F - bb.z;
        d = aa.w - uu.w; lp += -0.5f * d * d * expf(-ll.w) - 0.5f * ll.w - 0.5f * LOG2PI_F - bb.w;
        float ratio = expf(lp);
        float clipped = fminf(fmaxf(ratio, 1.0f - EPS_F), 1.0f + EPS_F);
        sAct += fminf(ratio * advn, clipped * advn);

        float dd = v[i] - retn;
        float ad = fabsf(dd);
        sCri += (ad < 1.0f) ? (0.5f * dd * dd) : (ad - 0.5f);
    }
    sAct = block_reduce_wmma(sAct, sred);
    sCri = block_reduce_wmma(sCri, sred);
    if (threadIdx.x == 0) { ws[WS_LACT + b] = sAct; ws[WS_LCRI + b] = sCri; }
}

// ---------------- K6: final scalar ----------------
__global__ void k6_final(const float* __restrict__ ws, float* __restrict__ out) {
    __shared__ float sred[NT];
    const int tid = threadIdx.x;
    float a = ws[WS_LACT + tid] + ws[WS_LACT + tid + NT];
    float c = ws[WS_LCRI + tid] + ws[WS_LCRI + tid + NT];
    a = block_reduce_wmma(a, sred);
    c = block_reduce_wmma(c, sred);
    if (tid == 0) {
        const float N = (float)T_TOTAL;
        out[0] = -(a / N) + (c / N);
    }
}

extern "C" void kernel_launch(void* const* d_in, const int* in_sizes, int n_in,
                              void* d_out, int out_size, void* d_ws, size_t ws_size,
                              hipStream_t stream) {
    const float* rewards = (const float*)d_in[0];
    const float* values  = (const float*)d_in[1];
    const float* beta    = (const float*)d_in[2];
    const float* actions = (const float*)d_in[3];
    const float* mu_pi   = (const float*)d_in[4];
    const float* lv_pi   = (const float*)d_in[5];
    const int*   masks   = (const int*)d_in[6];
    float* ws  = (float*)d_ws;
    float* out = (float*)d_out;

    k1_block_maps<<<NB, NT, 0, stream>>>(rewards, values, masks, ws);
    k2_block_scan<<<1, 32, 0, stream>>>(ws);
    k3_stats<<<NB, NT, 0, stream>>>(rewards, values, masks, ws);
    k4_stats_final<<<1, NT, 0, stream>>>(ws);
    k5_loss<<<NB, NT, 0, stream>>>(rewards, values, masks, beta, actions, mu_pi, lv_pi, ws);
    k6_final<<<1, NT, 0, stream>>>(ws, out);
}